// D_WC_3831110828814
// MI455X (gfx1250) — compile-verified
//
#include <hip/hip_runtime.h>

// Wilson-clover Dirac operator, LAT = (T,Z,Y,X) = (32,16,16,16), complex fp32.
// Kernel 1: precompute clover field strength F_{mu<nu}(x) (6 x 3x3 complex per site) into d_ws.
//           ws bytes needed: 6*131072*9*8 = 56,623,104.
// Kernel 2: per wave = 32 sites (two X-lines). 14 spin transforms (8 Wilson projectors,
//           6 * (-0.5*sigma)) done as V_WMMA_F32_16X16X4_F32 pairs (real/imag split),
//           staged via LDS. Neighbor-line staging uses GLOBAL_LOAD_ASYNC_TO_LDS_B128
//           (+ s_wait_asynccnt) when the toolchain exposes the gfx1250 builtins.

#define LT 32
#define LZ 16
#define LY 16
#define LX 16
#define VOL (LT*LZ*LY*LX)
#define WPB 2   // waves per block

#if defined(__has_builtin)
#if __has_builtin(__builtin_amdgcn_global_load_async_to_lds_b128)
#define HAVE_ASYNC_LDS 1
#endif
#endif
#ifndef HAVE_ASYNC_LDS
#define HAVE_ASYNC_LDS 0
#endif

typedef float v2f __attribute__((ext_vector_type(2)));
typedef float v8f __attribute__((ext_vector_type(8)));
typedef int   v4i __attribute__((ext_vector_type(4)));

// ---- 14 constant spin matrices, row-major M[i][j]; out = psi @ M ----
// t0..3 : -(G_k + I)/2   (forward hop, k = t+1)
// t4..7 :  (G_k - I)/2   (backward hop, k = t-3)
// t8..13: -0.5*sigma_{mu nu} for pairs (0,1),(0,2),(0,3),(1,2),(1,3),(2,3)
__constant__ float cMRE[14][16] = {
  {-0.5f,0,0,0,  0,-0.5f,0,0,  0,0,-0.5f,0,  0,0,0,-0.5f},
  {-0.5f,0,0,0.5f, 0,-0.5f,-0.5f,0, 0,-0.5f,-0.5f,0, 0.5f,0,0,-0.5f},
  {-0.5f,0,0,0,  0,-0.5f,0,0,  0,0,-0.5f,0,  0,0,0,-0.5f},
  {-0.5f,0,-0.5f,0, 0,-0.5f,0,-0.5f, -0.5f,0,-0.5f,0, 0,-0.5f,0,-0.5f},
  {-0.5f,0,0,0,  0,-0.5f,0,0,  0,0,-0.5f,0,  0,0,0,-0.5f},
  {-0.5f,0,0,-0.5f, 0,-0.5f,0.5f,0, 0,0.5f,-0.5f,0, -0.5f,0,0,-0.5f},
  {-0.5f,0,0,0,  0,-0.5f,0,0,  0,0,-0.5f,0,  0,0,0,-0.5f},
  {-0.5f,0,0.5f,0, 0,-0.5f,0,0.5f, 0.5f,0,-0.5f,0, 0,0.5f,0,-0.5f},
  {-0.5f,0,0,0, 0,0.5f,0,0, 0,0,-0.5f,0, 0,0,0,0.5f},
  {0},
  {0,0.5f,0,0, 0.5f,0,0,0, 0,0,0,-0.5f, 0,0,-0.5f,0},
  {0,-0.5f,0,0, -0.5f,0,0,0, 0,0,0,-0.5f, 0,0,-0.5f,0},
  {0},
  {0.5f,0,0,0, 0,-0.5f,0,0, 0,0,-0.5f,0, 0,0,0,0.5f},
};
__constant__ float cMIM[14][16] = {
  {0,0,0,-0.5f, 0,0,-0.5f,0, 0,0.5f,0,0, 0.5f,0,0,0},
  {0},
  {0,0,-0.5f,0, 0,0,0,0.5f, 0.5f,0,0,0, 0,-0.5f,0,0},
  {0},
  {0,0,0,0.5f, 0,0,0.5f,0, 0,-0.5f,0,0, -0.5f,0,0,0},
  {0},
  {0,0,0.5f,0, 0,0,0,-0.5f, -0.5f,0,0,0, 0,0.5f,0,0},
  {0},
  {0},
  {0,0.5f,0,0, -0.5f,0,0,0, 0,0,0,0.5f, 0,0,-0.5f,0},
  {0},
  {0},
  {0,0.5f,0,0, -0.5f,0,0,0, 0,0,0,-0.5f, 0,0,0.5f,0},
  {0},
};
__constant__ int cPMU[6] = {0,0,0,1,1,2};
__constant__ int cPNU[6] = {1,2,3,2,3,3};

__device__ __forceinline__ int siteIdx(int t,int z,int y,int x){ return ((t*LZ+z)*LY+y)*LX+x; }

__device__ __forceinline__ void shiftC(int&t,int&z,int&y,int&x,int axis,int s){
  switch(axis){
    case 0: t+=s; if(t<0)t+=LT; else if(t>=LT)t-=LT; break;
    case 1: z+=s; if(z<0)z+=LZ; else if(z>=LZ)z-=LZ; break;
    case 2: y+=s; if(y<0)y+=LY; else if(y>=LY)y-=LY; break;
    default:x+=s; if(x<0)x+=LX; else if(x>=LX)x-=LX; break;
  }
}

// ---------------- kernel 1: clover field strength ----------------
struct C3 { float re[9]; float im[9]; };

__device__ __forceinline__ void loadU(C3&M, const float* __restrict__ gre,
                                      const float* __restrict__ gim,
                                      int dir, int site, bool dag){
  const float* pr = gre + ((size_t)dir*VOL + site)*9;
  const float* pi = gim + ((size_t)dir*VOL + site)*9;
  #pragma unroll
  for(int a=0;a<3;a++){
    #pragma unroll
    for(int b=0;b<3;b++){
      int s_ = dag ? (b*3+a) : (a*3+b);
      float r = pr[s_], i = pi[s_];
      M.re[a*3+b] = r;
      M.im[a*3+b] = dag ? -i : i;
    }
  }
}

__device__ __forceinline__ C3 mulC3(const C3&A, const C3&B){
  C3 C;
  #pragma unroll
  for(int a=0;a<3;a++){
    #pragma unroll
    for(int b=0;b<3;b++){
      float cr=0.f, ci=0.f;
      #pragma unroll
      for(int k=0;k<3;k++){
        float ar=A.re[a*3+k], ai=A.im[a*3+k];
        float br=B.re[k*3+b], bi=B.im[k*3+b];
        cr = fmaf(ar,br,fmaf(-ai,bi,cr));
        ci = fmaf(ar,bi,fmaf( ai,br,ci));
      }
      C.re[a*3+b]=cr; C.im[a*3+b]=ci;
    }
  }
  return C;
}

__device__ __forceinline__ void addInto(C3&Q, const C3&P){
  #pragma unroll
  for(int e=0;e<9;e++){ Q.re[e]+=P.re[e]; Q.im[e]+=P.im[e]; }
}

__global__ void __launch_bounds__(256) clover_kernel(const float* __restrict__ gre,
                                                     const float* __restrict__ gim,
                                                     float2* __restrict__ cl){
  const int site = blockIdx.x*256 + threadIdx.x;
  const int p    = blockIdx.y;
  const int mu = cPMU[p], nu = cPNU[p];
  const int x = site & 15, y = (site>>4)&15, z = (site>>8)&15, t = site>>12;

  auto SC = [&](int ax,int sg,int ax2,int sg2)->int{
    int ct=t, cz=z, cy=y, cx=x;
    shiftC(ct,cz,cy,cx,ax,sg);
    if (ax2 >= 0) shiftC(ct,cz,cy,cx,ax2,sg2);
    return siteIdx(ct,cz,cy,cx);
  };
  const int s_m   = SC(mu,-1,-1,0);
  const int s_n   = SC(nu,-1,-1,0);
  const int s_mn  = SC(mu,-1,nu,-1);
  const int s_pm  = SC(mu,+1,-1,0);
  const int s_pn  = SC(nu,+1,-1,0);
  const int s_pmn = SC(mu,+1,nu,-1);
  const int s_mpn = SC(mu,-1,nu,+1);

  C3 Q;
  #pragma unroll
  for(int e=0;e<9;e++){ Q.re[e]=0.f; Q.im[e]=0.f; }

  C3 A,B,P;
  // M1 = Ud(nu,x-n) Ud(mu,x-m-n) U(nu,x-m-n) U(mu,x-m)
  loadU(A,gre,gim,nu,s_n, true );  loadU(B,gre,gim,mu,s_mn,true );  P = mulC3(A,B);
  loadU(B,gre,gim,nu,s_mn,false);  P = mulC3(P,B);
  loadU(B,gre,gim,mu,s_m, false);  P = mulC3(P,B);
  addInto(Q,P);
  // M2 = U(mu,x) Ud(nu,x+m-n) Ud(mu,x-n) U(nu,x-n)
  loadU(A,gre,gim,mu,site,false);  loadU(B,gre,gim,nu,s_pmn,true); P = mulC3(A,B);
  loadU(B,gre,gim,mu,s_n, true );  P = mulC3(P,B);
  loadU(B,gre,gim,nu,s_n, false);  P = mulC3(P,B);
  addInto(Q,P);
  // M3 = Ud(mu,x-m) U(nu,x-m) U(mu,x-m+n) Ud(nu,x)
  loadU(A,gre,gim,mu,s_m, true );  loadU(B,gre,gim,nu,s_m, false); P = mulC3(A,B);
  loadU(B,gre,gim,mu,s_mpn,false); P = mulC3(P,B);
  loadU(B,gre,gim,nu,site,true );  P = mulC3(P,B);
  addInto(Q,P);
  // M4 = U(nu,x) U(mu,x+n) Ud(nu,x+m) Ud(mu,x)
  loadU(A,gre,gim,nu,site,false);  loadU(B,gre,gim,mu,s_pn,false); P = mulC3(A,B);
  loadU(B,gre,gim,nu,s_pm,true );  P = mulC3(P,B);
  loadU(B,gre,gim,mu,site,true );  P = mulC3(P,B);
  addInto(Q,P);

  // F = 0.125 * (Q - Q^dagger)
  float2* dst = cl + ((size_t)p*VOL + site)*9;
  #pragma unroll
  for(int a=0;a<3;a++){
    #pragma unroll
    for(int b=0;b<3;b++){
      float fr = 0.125f*(Q.re[a*3+b] - Q.re[b*3+a]);
      float fi = 0.125f*(Q.im[a*3+b] + Q.im[b*3+a]);
      dst[a*3+b] = make_float2(fr, fi);
    }
  }
}

// ---------------- kernel 2: D_wc via WMMA spin transforms ----------------
__device__ __forceinline__ void bvals(int tt, int k, int n, float& o1, float& o2){
  // B1[k][n] = n<4 ? Mre[k][n] : Mim[k][n-4];  B2[k][n] = n<4 ? -Mim[k][n] : Mre[k][n-4]
  if (n >= 8){ o1 = 0.f; o2 = 0.f; return; }
  const int j = n & 3;
  const float re = cMRE[tt][k*4+j];
  const float im = cMIM[tt][k*4+j];
  if (n < 4){ o1 = re; o2 = -im; }
  else      { o1 = im; o2 =  re; }
}

__global__ void __launch_bounds__(32*WPB) dwc_kernel(
    const float* __restrict__ fre, const float* __restrict__ fim,
    const float* __restrict__ gre, const float* __restrict__ gim,
    const float2* __restrict__ cl, float* __restrict__ out)
{
  // staging LDS: per site raw [12 re | 12 im] floats (three 16B chunks per half,
  // matching GLOBAL_LOAD_ASYNC_TO_LDS_B128 granularity)
  __shared__ __align__(16) float sOwn[WPB][2][16][2][12];
  __shared__ __align__(16) float sNgb[WPB][2][16][2][12];
  // transformed spinors: [line][site][color][re/im][spin]
  __shared__ __align__(16) float sT  [WPB][2][16][3][2][4];

  const int lane = threadIdx.x & 31;
  const int w    = threadIdx.x >> 5;
  const int grp  = blockIdx.x*WPB + w;        // 32-site group (2 X-lines)
  const int site = grp*32 + lane;
  const int l    = lane >> 4;                 // line within group
  const int xr   = lane & 15;                 // x coordinate / tile row
  const int x = site & 15, y = (site>>4)&15, z = (site>>8)&15, t = site>>12;

  // prefetch this site's clover rows (resident in L2 after kernel 1)
  #pragma unroll
  for(int p=0;p<6;p++)
    __builtin_prefetch((const void*)(cl + ((size_t)p*VOL + site)*9), 0, 0);

  // own spinor: acc = (MASS+4) * psi, stage to LDS
  const float4* opr = (const float4*)(fre + (size_t)site*12);
  const float4* opi = (const float4*)(fim + (size_t)site*12);
  float4 pr0=opr[0], pr1=opr[1], pr2=opr[2];
  float4 pi0=opi[0], pi1=opi[1], pi2=opi[2];

  float __attribute__((aligned(16))) fr[12], fi2[12];
  *(float4*)&fr[0]=pr0;  *(float4*)&fr[4]=pr1;  *(float4*)&fr[8]=pr2;
  *(float4*)&fi2[0]=pi0; *(float4*)&fi2[4]=pi1; *(float4*)&fi2[8]=pi2;

  float accR[3][4], accI[3][4];
  #pragma unroll
  for(int c=0;c<3;c++){
    #pragma unroll
    for(int s=0;s<4;s++){ accR[c][s]=4.5f*fr[c*4+s]; accI[c][s]=4.5f*fi2[c*4+s]; }
  }
  {
    float4* so = (float4*)&sOwn[w][l][xr][0][0];
    so[0]=pr0; so[1]=pr1; so[2]=pr2;
    so[3]=pi0; so[4]=pi1; so[5]=pi2;
  }

  const int k0 = (lane < 16) ? 0 : 2;   // K indices held by this lane (A & B operands)
  const int nB = lane & 15;             // tile column for B/D

  for (int tt = 0; tt < 14; ++tt){
    // ---- stage source spinors ----
    int rshift = 0;
    bool useN = false;
    if (tt < 8){
      const int a   = tt & 3;                 // 0:T 1:Z 2:Y 3:X
      const int sgn = (tt < 4) ? -1 : 1;      // +dir hops read backward neighbor
      if (a == 3){
        rshift = sgn;                         // X hop: permute rows of own lines
      } else {
        int ct=t, cz=z, cy=y, cx=x;
        shiftC(ct,cz,cy,cx,a,sgn);
        const int ns = siteIdx(ct,cz,cy,cx);
#if HAVE_ASYNC_LDS
        // CDNA5 direct global->LDS DMA path (per-lane, ASYNCcnt-tracked)
        typedef __attribute__((address_space(1))) v4i* g4_ptr;
        typedef __attribute__((address_space(3))) v4i* l4_ptr;
        g4_ptr gr = (g4_ptr)(fre + (size_t)ns*12);
        g4_ptr gi = (g4_ptr)(fim + (size_t)ns*12);
        l4_ptr lr = (l4_ptr)&sNgb[w][l][xr][0][0];
        l4_ptr li = (l4_ptr)&sNgb[w][l][xr][1][0];
        __builtin_amdgcn_global_load_async_to_lds_b128(gr,   lr,   0, 0);
        __builtin_amdgcn_global_load_async_to_lds_b128(gr+1, lr+1, 0, 0);
        __builtin_amdgcn_global_load_async_to_lds_b128(gr+2, lr+2, 0, 0);
        __builtin_amdgcn_global_load_async_to_lds_b128(gi,   li,   0, 0);
        __builtin_amdgcn_global_load_async_to_lds_b128(gi+1, li+1, 0, 0);
        __builtin_amdgcn_global_load_async_to_lds_b128(gi+2, li+2, 0, 0);
#if __has_builtin(__builtin_amdgcn_s_wait_asynccnt)
        __builtin_amdgcn_s_wait_asynccnt(0);
#else
        asm volatile("s_wait_asynccnt 0x0" ::: "memory");
#endif
#else
        const float4* qr = (const float4*)(fre + (size_t)ns*12);
        const float4* qi = (const float4*)(fim + (size_t)ns*12);
        float4 a0=qr[0], a1=qr[1], a2=qr[2], b0=qi[0], b1q=qi[1], b2q=qi[2];
        float4* sn = (float4*)&sNgb[w][l][xr][0][0];
        sn[0]=a0; sn[1]=a1; sn[2]=a2;
        sn[3]=b0; sn[4]=b1q; sn[5]=b2q;
#endif
        useN = true;
      }
    }
    __builtin_amdgcn_wave_barrier();

    // ---- WMMA spin transform: T = psi_src @ M_tt (complex via 2 chained f32 WMMAs) ----
    float b1x,b1y,b2x,b2y;
    bvals(tt, k0,   nB, b1x, b2x);
    bvals(tt, k0+1, nB, b1y, b2y);
    v2f B1 = {b1x, b1y};
    v2f B2 = {b2x, b2y};
    const int rs = ((lane & 15) + rshift + 16) & 15;   // A row -> source site (X-permute)

    #pragma unroll
    for (int li2 = 0; li2 < 2; ++li2){
      #pragma unroll
      for (int c = 0; c < 3; ++c){
        const float* sp = useN ? &sNgb[w][li2][rs][0][0] : &sOwn[w][li2][rs][0][0];
        v2f Ar = { sp[c*4+k0],    sp[c*4+k0+1]    };
        v2f Ai = { sp[12+c*4+k0], sp[12+c*4+k0+1] };
        v8f D = {0.f,0.f,0.f,0.f,0.f,0.f,0.f,0.f};
        D = __builtin_amdgcn_wmma_f32_16x16x4_f32(false, Ar, false, B1, (short)0, D, false, false);
        D = __builtin_amdgcn_wmma_f32_16x16x4_f32(false, Ai, false, B2, (short)0, D, false, false);
        if (nB < 8){                                    // cols 0..3 = Re, 4..7 = Im
          const int ri = nB >> 2, ss = nB & 3;
          const int rbase = (lane < 16) ? 0 : 8;
          #pragma unroll
          for (int v = 0; v < 8; ++v)
            sT[w][li2][rbase + v][c][ri][ss] = D[v];
        }
      }
    }
    __builtin_amdgcn_wave_barrier();

    // ---- VALU: acc += Mat(site) @ T(site) ----
    float MR[3][3], MI[3][3];
    if (tt < 8){
      const int a = tt & 3;
      int ls; bool dag;
      if (tt < 4){
        int ct=t, cz=z, cy=y, cx=x; shiftC(ct,cz,cy,cx,a,-1);
        ls = siteIdx(ct,cz,cy,cx); dag = true;          // U_k(x-k)^dagger
      } else { ls = site; dag = false; }                // U_k(x)
      const float* lr = gre + ((size_t)a*VOL + ls)*9;
      const float* li9= gim + ((size_t)a*VOL + ls)*9;
      #pragma unroll
      for(int r=0;r<3;r++){
        #pragma unroll
        for(int c2=0;c2<3;c2++){
          const int s_ = dag ? (c2*3+r) : (r*3+c2);
          MR[r][c2] = lr[s_];
          const float ii = li9[s_];
          MI[r][c2] = dag ? -ii : ii;
        }
      }
    } else {
      const float2* fp = cl + ((size_t)(tt-8)*VOL + site)*9;
      #pragma unroll
      for(int r=0;r<3;r++){
        #pragma unroll
        for(int c2=0;c2<3;c2++){ float2 v = fp[r*3+c2]; MR[r][c2]=v.x; MI[r][c2]=v.y; }
      }
    }

    float __attribute__((aligned(16))) Tr[12], Ti[12];
    {
      const float4* tp = (const float4*)&sT[w][l][xr][0][0][0];
      *(float4*)&Tr[0]=tp[0]; *(float4*)&Ti[0]=tp[1];
      *(float4*)&Tr[4]=tp[2]; *(float4*)&Ti[4]=tp[3];
      *(float4*)&Tr[8]=tp[4]; *(float4*)&Ti[8]=tp[5];
    }
    #pragma unroll
    for(int r=0;r<3;r++){
      #pragma unroll
      for(int s=0;s<4;s++){
        float ar=accR[r][s], ai=accI[r][s];
        #pragma unroll
        for(int c2=0;c2<3;c2++){
          const float tr=Tr[c2*4+s], ti=Ti[c2*4+s];
          ar = fmaf(MR[r][c2],tr, fmaf(-MI[r][c2],ti, ar));
          ai = fmaf(MR[r][c2],ti, fmaf( MI[r][c2],tr, ai));
        }
        accR[r][s]=ar; accI[r][s]=ai;
      }
    }
    __builtin_amdgcn_wave_barrier();
  }

  // store: out[..., 3,4,2] (re,im interleaved)
  float2* op = (float2*)out;
  #pragma unroll
  for(int c=0;c<3;c++){
    #pragma unroll
    for(int s=0;s<4;s++)
      op[(size_t)site*12 + c*4 + s] = make_float2(accR[c][s], accI[c][s]);
  }
}

extern "C" void kernel_launch(void* const* d_in, const int* in_sizes, int n_in,
                              void* d_out, int out_size, void* d_ws, size_t ws_size,
                              hipStream_t stream) {
  const float* fre = (const float*)d_in[0];
  const float* fim = (const float*)d_in[1];
  const float* gre = (const float*)d_in[2];
  const float* gim = (const float*)d_in[3];
  float*  out = (float*)d_out;
  float2* cl  = (float2*)d_ws;   // needs 6*VOL*9*sizeof(float2) = 56,623,104 bytes

  dim3 gridC(VOL/256, 6, 1);
  clover_kernel<<<gridC, 256, 0, stream>>>(gre, gim, cl);
  dwc_kernel<<<VOL/(32*WPB), 32*WPB, 0, stream>>>(fre, fim, gre, gim, cl, out);

  (void)in_sizes; (void)n_in; (void)out_size; (void)ws_size;
}